// Voxelization_88467736363821
// MI455X (gfx1250) — compile-verified
//
#include <hip/hip_runtime.h>
#include <cstdint>
#include <cstddef>

// Problem constants (from reference)
#define B_   16
#define C_   64
#define N_   65536
#define R_   32768      // 32*32*32
#define NB_  64         // reduction blocks per batch

// ---------------- CDNA5 async global->LDS helpers --------------------------
// Per-lane DMA of one dword from global memory into LDS, tracked by ASYNCcnt.
__device__ __forceinline__ void async_ld_b32(const float* gptr, float* lptr) {
#if __has_builtin(__builtin_amdgcn_global_load_async_to_lds_b32)
  __builtin_amdgcn_global_load_async_to_lds_b32(
      (__attribute__((address_space(1))) int*)gptr,
      (__attribute__((address_space(3))) int*)lptr,
      0, 0);
#else
  unsigned loff = (unsigned)(uintptr_t)(__attribute__((address_space(3))) float*)lptr;
  asm volatile("global_load_async_to_lds_b32 %0, %1, off"
               :: "v"(loff), "v"(gptr) : "memory");
#endif
}

__device__ __forceinline__ void wait_async_1() {
#if __has_builtin(__builtin_amdgcn_s_wait_asynccnt)
  __builtin_amdgcn_s_wait_asynccnt(1);
#else
  asm volatile("s_wait_asynccnt 1" ::: "memory");
#endif
}
__device__ __forceinline__ void wait_async_0() {
#if __has_builtin(__builtin_amdgcn_s_wait_asynccnt)
  __builtin_amdgcn_s_wait_asynccnt(0);
#else
  asm volatile("s_wait_asynccnt 0" ::: "memory");
#endif
}

__device__ __forceinline__ void gatomic_fadd(float* p, float v) {
  // Native global_atomic_add_f32 (no CAS loop) — accumulator lives in L2.
  unsafeAtomicAdd(p, v);
}

// ---------------- Kernel 5 (defined first so disasm snippet shows it):
// normalize coords + scatter-add features.
// Each thread owns one point. The 64 channel values (stride-N apart in
// memory, fully coalesced across a wave) are streamed through LDS with a
// double-buffered CDNA5 async global->LDS pipeline: channel c+1 is in flight
// while channel c's scattered L2 atomic is issued.
__global__ void __launch_bounds__(256)
k_scatter(const float* __restrict__ feats, const float* __restrict__ coords,
          const float* __restrict__ mean, const unsigned* __restrict__ maxn,
          float* __restrict__ out_vox, float* __restrict__ out_nc,
          float* __restrict__ cnt) {
  __shared__ float tile[2][256];
  const int b = blockIdx.y, tid = threadIdx.x;
  const int n = blockIdx.x * 256 + tid;

  const float denom = __uint_as_float(maxn[b]) * 2.0f; // EPS == 0
  const float m0 = mean[b * 3 + 0], m1 = mean[b * 3 + 1], m2 = mean[b * 3 + 2];
  const float* cb = coords + (size_t)b * 3 * N_;

  float nc0 = (cb[n]          - m0) / denom + 0.5f;
  float nc1 = (cb[N_ + n]     - m1) / denom + 0.5f;
  float nc2 = (cb[2 * N_ + n] - m2) / denom + 0.5f;
  nc0 = fminf(fmaxf(nc0 * 32.0f, 0.0f), 31.0f);
  nc1 = fminf(fmaxf(nc1 * 32.0f, 0.0f), 31.0f);
  nc2 = fminf(fmaxf(nc2 * 32.0f, 0.0f), 31.0f);

  float* onc = out_nc + (size_t)b * 3 * N_;
  onc[n]          = nc0;
  onc[N_ + n]     = nc1;
  onc[2 * N_ + n] = nc2;

  const int v0 = (int)rintf(nc0);                 // RNE matches jnp.round
  const int v1 = (int)rintf(nc1);
  const int v2 = (int)rintf(nc2);
  const int idx = v0 * 1024 + v1 * 32 + v2;

  gatomic_fadd(cnt + (size_t)b * R_ + idx, 1.0f);

  const float* fb = feats + (size_t)b * C_ * N_ + n;
  float* ob = out_vox + (size_t)b * C_ * R_ + idx;

  async_ld_b32(fb, &tile[0][tid]);                // prime channel 0
  for (int c = 0; c < C_; ++c) {
    if (c + 1 < C_) {
      async_ld_b32(fb + (size_t)(c + 1) * N_, &tile[(c + 1) & 1][tid]);
      wait_async_1();                             // current channel landed
    } else {
      wait_async_0();
    }
    float v = *(volatile float*)&tile[c & 1][tid];
    gatomic_fadd(ob + (size_t)c * R_, v);
  }
}

// ---------------- Kernel 1: zero accumulator + counts + max slots ----------
__global__ void __launch_bounds__(256)
k_init(float4* __restrict__ vox4, float4* __restrict__ cnt4,
       unsigned* __restrict__ maxn, int nv4, int nc4) {
  const float4 z = make_float4(0.f, 0.f, 0.f, 0.f);
  int stride = gridDim.x * blockDim.x;
  for (int i = blockIdx.x * blockDim.x + threadIdx.x; i < nv4 + nc4; i += stride) {
    if (i < nv4) vox4[i] = z;
    else         cnt4[i - nv4] = z;
  }
  if (blockIdx.x == 0 && threadIdx.x < B_) maxn[threadIdx.x] = 0u;
}

// ---------------- Kernel 2: per-block partial sums of coords ---------------
__global__ void __launch_bounds__(256)
k_mean_partial(const float* __restrict__ coords, float* __restrict__ partial) {
  __shared__ float sx[256], sy[256], sz[256];
  const int b = blockIdx.y, blk = blockIdx.x, tid = threadIdx.x;
  const int PPB = N_ / NB_;                       // 1024 points per block
  const float* cb = coords + (size_t)b * 3 * N_;
  float ax = 0.f, ay = 0.f, az = 0.f;
  int n0 = blk * PPB + tid;
  #pragma unroll
  for (int i = 0; i < PPB / 256; ++i) {
    int n = n0 + i * 256;
    ax += cb[n];
    ay += cb[N_ + n];
    az += cb[2 * N_ + n];
  }
  sx[tid] = ax; sy[tid] = ay; sz[tid] = az;
  __syncthreads();
  for (int s = 128; s > 0; s >>= 1) {
    if (tid < s) {
      sx[tid] += sx[tid + s];
      sy[tid] += sy[tid + s];
      sz[tid] += sz[tid + s];
    }
    __syncthreads();
  }
  if (tid == 0) {
    float* p = partial + ((size_t)b * NB_ + blk) * 3;
    p[0] = sx[0]; p[1] = sy[0]; p[2] = sz[0];
  }
}

// ---------------- Kernel 3: combine partials -> mean (fixed order) ---------
__global__ void __launch_bounds__(64)
k_mean_final(const float* __restrict__ partial, float* __restrict__ mean) {
  __shared__ float sx[64], sy[64], sz[64];
  const int b = blockIdx.x, tid = threadIdx.x;
  const float* p = partial + ((size_t)b * NB_ + tid) * 3;
  sx[tid] = p[0]; sy[tid] = p[1]; sz[tid] = p[2];
  __syncthreads();
  for (int s = 32; s > 0; s >>= 1) {
    if (tid < s) {
      sx[tid] += sx[tid + s];
      sy[tid] += sy[tid + s];
      sz[tid] += sz[tid + s];
    }
    __syncthreads();
  }
  if (tid == 0) {
    const float inv = 1.0f / (float)N_;           // exact power of two
    mean[b * 3 + 0] = sx[0] * inv;
    mean[b * 3 + 1] = sy[0] * inv;
    mean[b * 3 + 2] = sz[0] * inv;
  }
}

// ---------------- Kernel 4: max point norm per batch -----------------------
__global__ void __launch_bounds__(256)
k_maxnorm(const float* __restrict__ coords, const float* __restrict__ mean,
          unsigned* __restrict__ maxn) {
  __shared__ float sm[256];
  const int b = blockIdx.y, blk = blockIdx.x, tid = threadIdx.x;
  const float mx = mean[b * 3 + 0], my = mean[b * 3 + 1], mz = mean[b * 3 + 2];
  const float* cb = coords + (size_t)b * 3 * N_;
  const int PPB = N_ / NB_;
  float best = 0.f;
  int n0 = blk * PPB + tid;
  #pragma unroll
  for (int i = 0; i < PPB / 256; ++i) {
    int n = n0 + i * 256;
    float dx = cb[n] - mx;
    float dy = cb[N_ + n] - my;
    float dz = cb[2 * N_ + n] - mz;
    best = fmaxf(best, dx * dx + dy * dy + dz * dz);
  }
  sm[tid] = best;
  __syncthreads();
  for (int s = 128; s > 0; s >>= 1) {
    if (tid < s) sm[tid] = fmaxf(sm[tid], sm[tid + s]);
    __syncthreads();
  }
  // non-negative floats: uint compare == float compare; exact & order-free
  if (tid == 0) atomicMax(maxn + b, __float_as_uint(sqrtf(sm[0])));
}

// ---------------- Kernel 6: divide sums by counts in place -----------------
__global__ void __launch_bounds__(256)
k_finalize(float4* __restrict__ out4, const float4* __restrict__ cnt4) {
  const int R4 = R_ / 4;
  const int total = B_ * C_ * R4;
  int stride = gridDim.x * blockDim.x;
  for (int i = blockIdx.x * blockDim.x + threadIdx.x; i < total; i += stride) {
    int r4 = i % R4;
    int b  = (i / R4) / C_;
    float4 s = out4[i];
    float4 c = cnt4[(size_t)b * R4 + r4];         // 2 MB, L2 resident
    s.x /= fmaxf(c.x, 1.0f);
    s.y /= fmaxf(c.y, 1.0f);
    s.z /= fmaxf(c.z, 1.0f);
    s.w /= fmaxf(c.w, 1.0f);
    out4[i] = s;
  }
}

// ---------------------------------------------------------------------------
extern "C" void kernel_launch(void* const* d_in, const int* in_sizes, int n_in,
                              void* d_out, int out_size, void* d_ws, size_t ws_size,
                              hipStream_t stream) {
  (void)in_sizes; (void)n_in; (void)out_size; (void)ws_size;
  const float* feats  = (const float*)d_in[0];   // [B, C, N]
  const float* coords = (const float*)d_in[1];   // [B, 3, N]

  float* out_vox = (float*)d_out;                          // [B, C, R]
  float* out_nc  = (float*)d_out + (size_t)B_ * C_ * R_;   // [B, 3, N]

  // workspace layout (~2.1 MB)
  float*    cnt     = (float*)d_ws;                        // B*R
  float*    partial = cnt + (size_t)B_ * R_;               // B*NB*3
  float*    mean    = partial + (size_t)B_ * NB_ * 3;      // B*3
  unsigned* maxn    = (unsigned*)(mean + B_ * 3);          // B

  const int nv4 = B_ * C_ * R_ / 4;
  const int nc4 = B_ * R_ / 4;

  k_init<<<4096, 256, 0, stream>>>((float4*)out_vox, (float4*)cnt, maxn, nv4, nc4);
  k_mean_partial<<<dim3(NB_, B_), 256, 0, stream>>>(coords, partial);
  k_mean_final<<<B_, 64, 0, stream>>>(partial, mean);
  k_maxnorm<<<dim3(NB_, B_), 256, 0, stream>>>(coords, mean, maxn);
  k_scatter<<<dim3(N_ / 256, B_), 256, 0, stream>>>(feats, coords, mean, maxn,
                                                    out_vox, out_nc, cnt);
  k_finalize<<<4096, 256, 0, stream>>>((float4*)out_vox, (const float4*)cnt);
}